// GraphCNN_59090160059061
// MI455X (gfx1250) — compile-verified
//
#include <hip/hip_runtime.h>

// ---------------- problem constants (match reference) ----------------
#define NNODES   100000
#define NEDGES   1600000
#define NGRAPHS  256
#define DH       128
#define DOUT     10
#define NLAYERS  4
#define BN_EPS   1e-5f

typedef float v2f __attribute__((ext_vector_type(2)));
typedef float v8f __attribute__((ext_vector_type(8)));

// ---------------------------------------------------------------------
// pooled = (1 + eps[l]) * h          (float4 vectorized, NNODES*32 threads)
// ---------------------------------------------------------------------
__global__ __launch_bounds__(256) void k_init_pooled(const float* __restrict__ h,
                                                     float* __restrict__ pooled,
                                                     const float* __restrict__ eps,
                                                     int l) {
    size_t tid = (size_t)blockIdx.x * 256 + threadIdx.x;   // exactly NNODES*32
    float sc = 1.0f + eps[l];
    float4 v = ((const float4*)h)[tid];
    float4 o;
    o.x = v.x * sc; o.y = v.y * sc; o.z = v.z * sc; o.w = v.w * sc;
    ((float4*)pooled)[tid] = o;
}

// ---------------------------------------------------------------------
// pooled[dst] += h[src]   (1 wave per edge, float4 gather + 4 f32 atomics)
// ---------------------------------------------------------------------
__global__ __launch_bounds__(256) void k_edge_scatter(const float* __restrict__ h,
                                                      const int* __restrict__ esrc,
                                                      const int* __restrict__ edst,
                                                      float* __restrict__ pooled) {
    size_t tid = (size_t)blockIdx.x * 256 + threadIdx.x;   // exactly NEDGES*32
    int e    = (int)(tid >> 5);
    int lane = (int)(tid & 31);
    int s = esrc[e];            // uniform across wave -> scalar load
    int d = edst[e];
    float4 v = *(const float4*)(h + (size_t)s * DH + lane * 4);
    float* p = pooled + (size_t)d * DH + lane * 4;
    unsafeAtomicAdd(p + 0, v.x);
    unsafeAtomicAdd(p + 1, v.y);
    unsafeAtomicAdd(p + 2, v.z);
    unsafeAtomicAdd(p + 3, v.w);
}

// ---------------------------------------------------------------------
// C[M x 128] = A[M x 128] @ W[128 x 128] + bias
// block = 256 threads = 8 waves; block owns 16 rows, wave w owns cols 16w..16w+15
// A tile staged in LDS (stride 132 floats -> conflict-free ds_load_b64 frags)
// V_WMMA_F32_16X16X4_F32, 32 MACs per wave over K=128.
// ---------------------------------------------------------------------
#define LDS_STRIDE 132
__global__ __launch_bounds__(256) void k_gemm128(const float* __restrict__ A,
                                                 const float* __restrict__ W,
                                                 const float* __restrict__ bias,
                                                 float* __restrict__ C) {
    __shared__ float As[16 * LDS_STRIDE];
    const int m0 = blockIdx.x * 16;
    const int t  = threadIdx.x;

    // cooperative stage of 16x128 A tile (8 floats per thread)
    {
        int r = t >> 4;               // 0..15
        int c = (t & 15) * 8;         // 0..120
        const float* src = A + (size_t)(m0 + r) * DH + c;
        float* dst = As + r * LDS_STRIDE + c;
#pragma unroll
        for (int i = 0; i < 8; ++i) dst[i] = src[i];
    }
    __syncthreads();

    const int wave = t >> 5;          // 0..7  -> N tile
    const int lane = t & 31;
    const int n0   = wave * 16;
    const int lo   = lane & 15;       // row of A frag / col of B frag
    const int hi   = lane >> 4;       // K-half selector (0: K0,K1  1: K2,K3)

    v8f acc = {};
    const float* Wc = W + n0 + lo;    // column base in row-major W[k][n]

#pragma unroll 8
    for (int k0 = 0; k0 < DH; k0 += 4) {
        int kk = k0 + (hi << 1);
        // A 16x4 frag: lanes 0-15 -> {K=k0,k0+1}; lanes 16-31 -> {K=k0+2,k0+3}
        v2f a = *(const v2f*)(As + lo * LDS_STRIDE + kk);
        // B 4x16 frag: mirrored K split across lane halves, N striped in lanes
        v2f b;
        b.x = Wc[(size_t)kk * DH];
        b.y = Wc[(size_t)(kk + 1) * DH];
        acc = __builtin_amdgcn_wmma_f32_16x16x4_f32(
            /*neg_a=*/false, a, /*neg_b=*/false, b,
            /*c_mod=*/(short)0, acc, /*reuse_a=*/false, /*reuse_b=*/false);
    }

    // C layout: VGPR r, lanes 0-15 -> M=r ; lanes 16-31 -> M=8+r ; N = lane&15
    float bc = bias[n0 + lo];
    float* Crow = C + (size_t)(m0 + hi * 8) * DH + n0 + lo;
#pragma unroll
    for (int r = 0; r < 8; ++r)
        Crow[(size_t)r * DH] = acc[r] + bc;
}

// ---------------------------------------------------------------------
// zero helper
// ---------------------------------------------------------------------
__global__ __launch_bounds__(256) void k_zero(float* __restrict__ p, int n) {
    int i = blockIdx.x * 256 + threadIdx.x;
    if (i < n) p[i] = 0.0f;
}

// ---------------------------------------------------------------------
// per-column sum / sumsq over NNODES rows -> stats[0..127]=sum, [128..255]=sumsq
// 200 blocks x 500 rows; two threads per column per block
// ---------------------------------------------------------------------
__global__ __launch_bounds__(256) void k_bn_stats(const float* __restrict__ X,
                                                  float* __restrict__ stats) {
    int col  = threadIdx.x & (DH - 1);
    int half = threadIdx.x >> 7;             // 0 or 1
    int base = blockIdx.x * 500;
    float s = 0.0f, q = 0.0f;
    for (int r = base + half; r < base + 500; r += 2) {
        float v = X[(size_t)r * DH + col];
        s += v;
        q += v * v;
    }
    unsafeAtomicAdd(&stats[col], s);
    unsafeAtomicAdd(&stats[DH + col], q);
}

// ---------------------------------------------------------------------
// Y = relu((X - mean) * rsqrt(var+eps) * gamma + beta)
// mode 0: Y only;  mode 1: fused = w*Y;  mode 2: fused += w*Y
// ---------------------------------------------------------------------
__global__ __launch_bounds__(256) void k_bn_apply(const float* __restrict__ X,
                                                  const float* __restrict__ stats,
                                                  const float* __restrict__ gamma,
                                                  const float* __restrict__ beta,
                                                  float* __restrict__ Y,
                                                  float* __restrict__ fused,
                                                  const float* __restrict__ fw,
                                                  int l, int mode) {
    size_t tid = (size_t)blockIdx.x * 256 + threadIdx.x;   // NNODES*32
    int c0 = (int)(tid & 31) * 4;
    const float invN = 1.0f / (float)NNODES;
    float4 v = ((const float4*)X)[tid];
    float r[4] = { v.x, v.y, v.z, v.w };
    float o[4];
#pragma unroll
    for (int j = 0; j < 4; ++j) {
        int c = c0 + j;
        float m   = stats[c] * invN;
        float var = stats[DH + c] * invN - m * m;
        float sc  = gamma[c] * rsqrtf(var + BN_EPS);
        float tv  = (r[j] - m) * sc + beta[c];
        o[j] = tv > 0.0f ? tv : 0.0f;
    }
    float4 y = { o[0], o[1], o[2], o[3] };
    ((float4*)Y)[tid] = y;
    if (mode) {
        float w = fw[l];
        float4 f;
        if (mode == 2) {
            f = ((float4*)fused)[tid];
            f.x += w * o[0]; f.y += w * o[1]; f.z += w * o[2]; f.w += w * o[3];
        } else {
            f.x = w * o[0]; f.y = w * o[1]; f.z = w * o[2]; f.w = w * o[3];
        }
        ((float4*)fused)[tid] = f;
    }
}

// ---------------------------------------------------------------------
// G[graph_ids[n]] += fused[n]    (atomics into 256x128 L2-resident buffer)
// ---------------------------------------------------------------------
__global__ __launch_bounds__(256) void k_graph_pool(const float* __restrict__ fused,
                                                    const int* __restrict__ gid,
                                                    float* __restrict__ G) {
    size_t tid = (size_t)blockIdx.x * 256 + threadIdx.x;   // NNODES*32
    int node = (int)(tid >> 5);
    int c    = (int)(tid & 31) * 4;
    int g = gid[node];                                     // uniform per wave
    float4 v = ((const float4*)fused)[tid];
    float* p = G + (size_t)g * DH + c;
    unsafeAtomicAdd(p + 0, v.x);
    unsafeAtomicAdd(p + 1, v.y);
    unsafeAtomicAdd(p + 2, v.z);
    unsafeAtomicAdd(p + 3, v.w);
}

// ---------------------------------------------------------------------
// out[256 x 10] = G[256 x 128] @ Wp[128 x 10] + bp   (tiny, scalar)
// ---------------------------------------------------------------------
__global__ __launch_bounds__(256) void k_final(const float* __restrict__ G,
                                               const float* __restrict__ Wp,
                                               const float* __restrict__ bp,
                                               float* __restrict__ out) {
    int idx = blockIdx.x * 256 + threadIdx.x;
    if (idx >= NGRAPHS * DOUT) return;
    int g = idx / DOUT, o = idx % DOUT;
    float s = bp[o];
    const float* row = G + (size_t)g * DH;
#pragma unroll 8
    for (int k = 0; k < DH; ++k)
        s += row[k] * Wp[(size_t)k * DOUT + o];
    out[idx] = s;
}

// ---------------------------------------------------------------------
extern "C" void kernel_launch(void* const* d_in, const int* in_sizes, int n_in,
                              void* d_out, int out_size, void* d_ws, size_t ws_size,
                              hipStream_t stream) {
    (void)in_sizes; (void)n_in; (void)out_size; (void)ws_size;
    const float* x    = (const float*)d_in[0];
    const int*   esrc = (const int*)  d_in[1];
    const int*   edst = (const int*)  d_in[2];
    const int*   gids = (const int*)  d_in[3];
    const float* eps  = (const float*)d_in[4];
    const float* fw   = (const float*)d_in[5];
    const float* W1   = (const float*)d_in[6];
    const float* b1   = (const float*)d_in[7];
    const float* g1   = (const float*)d_in[8];
    const float* bt1  = (const float*)d_in[9];
    const float* W2   = (const float*)d_in[10];
    const float* b2   = (const float*)d_in[11];
    const float* g2   = (const float*)d_in[12];
    const float* bt2  = (const float*)d_in[13];
    const float* Wp   = (const float*)d_in[14];
    const float* bp   = (const float*)d_in[15];
    float* out = (float*)d_out;

    char* ws = (char*)d_ws;
    const size_t SZ = (size_t)NNODES * DH * sizeof(float);   // 51.2 MB
    float* pooled = (float*)(ws);
    float* tbuf   = (float*)(ws + SZ);
    float* hbuf   = (float*)(ws + 2 * SZ);
    float* fused  = (float*)(ws + 3 * SZ);
    float* stats  = (float*)(ws + 4 * SZ);      // 256 floats
    float* G      = stats + 2 * DH;             // 256*128 floats (1 KB aligned)

    const int nodeBlocks = (NNODES * 32) / 256;     // 12500 (exact)
    const int edgeBlocks = (NEDGES * 32) / 256;     // 200000 (exact)
    const int gemmBlocks = NNODES / 16;             // 6250 (exact)

    const float* hin = x;
    for (int l = 0; l < NLAYERS; ++l) {
        // aggregation: pooled = (1+eps)*h + sum_{e: dst=v} h[src(e)]
        k_init_pooled<<<nodeBlocks, 256, 0, stream>>>(hin, pooled, eps, l);
        k_edge_scatter<<<edgeBlocks, 256, 0, stream>>>(hin, esrc, edst, pooled);
        // MLP layer 1: GEMM -> BN -> ReLU
        k_gemm128<<<gemmBlocks, 256, 0, stream>>>(pooled, W1 + (size_t)l * DH * DH,
                                                  b1 + l * DH, tbuf);
        k_zero<<<1, 256, 0, stream>>>(stats, 2 * DH);
        k_bn_stats<<<200, 256, 0, stream>>>(tbuf, stats);
        k_bn_apply<<<nodeBlocks, 256, 0, stream>>>(tbuf, stats, g1 + l * DH,
                                                   bt1 + l * DH, tbuf, nullptr,
                                                   fw, l, 0);
        // MLP layer 2: GEMM -> BN -> ReLU (+ weighted fusion accumulation)
        k_gemm128<<<gemmBlocks, 256, 0, stream>>>(tbuf, W2 + (size_t)l * DH * DH,
                                                  b2 + l * DH, pooled);
        k_zero<<<1, 256, 0, stream>>>(stats, 2 * DH);
        k_bn_stats<<<200, 256, 0, stream>>>(pooled, stats);
        k_bn_apply<<<nodeBlocks, 256, 0, stream>>>(pooled, stats, g2 + l * DH,
                                                   bt2 + l * DH, hbuf, fused,
                                                   fw, l, (l == 0) ? 1 : 2);
        hin = hbuf;
    }
    // graph sum pooling + final projection
    k_zero<<<(NGRAPHS * DH) / 256, 256, 0, stream>>>(G, NGRAPHS * DH);
    k_graph_pool<<<nodeBlocks, 256, 0, stream>>>(fused, gids, G);
    k_final<<<(NGRAPHS * DOUT + 255) / 256, 256, 0, stream>>>(G, Wp, bp, out);
}